// GNN_model_9002251452616
// MI455X (gfx1250) — compile-verified
//
#include <hip/hip_runtime.h>
#include <cstdint>

typedef __attribute__((ext_vector_type(2))) float v2f;
typedef __attribute__((ext_vector_type(8))) float v8f;

#define NN 50000
#define NE 800000
#define NG 500
#define F 100
#define KPAD 36
#define BN_EPS 1e-5f

// ---------------- degree / norm ----------------
__global__ void init_ones(float* p, int n) {
    int t = blockIdx.x * blockDim.x + threadIdx.x;
    if (t < n) p[t] = 1.0f;  // self-loop contributes 1 to every node's degree
}

__global__ void deg_accum(const long long* __restrict__ dst, float* __restrict__ deg, int e) {
    int t = blockIdx.x * blockDim.x + threadIdx.x;
    if (t < e) atomicAdd(&deg[dst[t]], 1.0f);
}

__global__ void deg_to_dinv(float* __restrict__ deg, int n) {
    int t = blockIdx.x * blockDim.x + threadIdx.x;
    if (t < n) {
        float d = deg[t];
        deg[t] = d > 0.f ? rsqrtf(d) : 0.f;
    }
}

// ---------------- pad x: [NN,33] -> [NN,36] (zero-filled K tail) ----------------
__global__ void pad_x(const float* __restrict__ x, float* __restrict__ xp) {
    const int i = blockIdx.x;
    const int f = threadIdx.x;  // 64 threads
    if (f < KPAD) xp[(size_t)i * KPAD + f] = (f < 33) ? x[(size_t)i * 33 + f] : 0.f;
}

// ---------------- fp32 WMMA GEMM: C[M,N] = A[M,K] @ B[K,N] (+bias)(+leaky) ----
// One wave per 16x16 output tile. Requires K % 4 == 0 and rows of A 8B-aligned
// (K even). Out-of-range lanes CLAMP their load indices (results for padded
// rows/cols are garbage but never stored), so all loads are unconditional:
// no exec masking, no per-element branches in the inner loop.
__global__ void wmma_gemm_f32(const float* __restrict__ A, const float* __restrict__ B,
                              float* __restrict__ C, int M, int N, int K,
                              const float* __restrict__ bias, int leaky) {
    const int lane = threadIdx.x & 31;
    const int wave = threadIdx.x >> 5;
    const int ntN = (N + 15) >> 4;
    const int ntM = (M + 15) >> 4;
    const int tile = blockIdx.x * (blockDim.x >> 5) + wave;
    if (tile >= ntN * ntM) return;  // wave-uniform: EXEC all-1s at every WMMA
    const int tm = tile / ntN, tn = tile % ntN;
    const int m0 = tm << 4, n0 = tn << 4;
    const int half = lane >> 4;  // 0 -> K pair {0,1}, 1 -> K pair {2,3}
    const int l15  = lane & 15;
    const int mA = m0 + l15;  // logical A row
    const int nB = n0 + l15;  // logical B/C column
    const int mAc = mA < M ? mA : M - 1;  // clamped (address-safe) indices
    const int nBc = nB < N ? nB : N - 1;
    const float* __restrict__ Arow = A + (size_t)mAc * K + (half << 1);
    const float* __restrict__ Bcol = B + nBc + (size_t)(half << 1) * N;
    v8f acc = {};
#pragma unroll 4
    for (int k0 = 0; k0 < K; k0 += 4) {
        const float2 av = *(const float2*)(Arow + k0);  // 8B-aligned contiguous pair
        v2f a, b;
        a.x = av.x;
        a.y = av.y;
        b.x = Bcol[(size_t)k0 * N];
        b.y = Bcol[(size_t)k0 * N + N];
        acc = __builtin_amdgcn_wmma_f32_16x16x4_f32(false, a, false, b,
                                                    (short)0, acc, false, false);
    }
    const int mBase = m0 + (half << 3);
#pragma unroll
    for (int v = 0; v < 8; ++v) {
        const int m = mBase + v;
        if (m < M && nB < N) {
            float val = acc[v];
            if (bias) val += bias[nB];
            if (leaky) val = val > 0.f ? val : 0.1f * val;
            C[(size_t)m * N + nB] = val;
        }
    }
}

// ---------------- edge scatter: agg[dst] += dinv[src]*dinv[dst] * m[src] ----
// 128 threads = 4 waves = 4 edges per block; float4 gather (row stride 400B)
__global__ void edge_scatter(const long long* __restrict__ src, const long long* __restrict__ dst,
                             const float* __restrict__ dinv, const float* __restrict__ m,
                             float* __restrict__ agg, int e) {
    const int lane = threadIdx.x & 31;
    const int sub  = threadIdx.x >> 5;
    const int ei   = blockIdx.x * 4 + sub;
    if (ei >= e) return;
    const long long s = src[ei];
    const long long d = dst[ei];
    const float w = dinv[s] * dinv[d];
    if (lane < F / 4) {
        const float4 v = ((const float4*)(m + (size_t)s * F))[lane];
        float* a = agg + (size_t)d * F + lane * 4;
        atomicAdd(a + 0, w * v.x);
        atomicAdd(a + 1, w * v.y);
        atomicAdd(a + 2, w * v.z);
        atomicAdd(a + 3, w * v.w);
    }
}

// ---------------- self-loop term + layer bias ----------------
__global__ void selfloop_bias(const float* __restrict__ dinv, const float* __restrict__ m,
                              const float* __restrict__ bias, float* __restrict__ agg) {
    const int i = blockIdx.x;
    const int f = threadIdx.x;
    if (f < F) {
        const float di = dinv[i];
        agg[(size_t)i * F + f] += di * di * m[(size_t)i * F + f] + bias[f];
    }
}

// ---------------- BN column statistics -> scale/shift ----------------
__global__ void bn_stats(const float* __restrict__ agg, const float* __restrict__ gamma,
                         const float* __restrict__ beta, float* __restrict__ scale,
                         float* __restrict__ shift, int n) {
    __shared__ float ssum[256];
    __shared__ float ssq[256];
    const int f = blockIdx.x;  // 0..F-1
    float s = 0.f, q = 0.f;
    for (int i = threadIdx.x; i < n; i += blockDim.x) {
        const float v = agg[(size_t)i * F + f];
        s += v;
        q += v * v;
    }
    ssum[threadIdx.x] = s;
    ssq[threadIdx.x]  = q;
    __syncthreads();
    for (int st = 128; st > 0; st >>= 1) {
        if (threadIdx.x < st) {
            ssum[threadIdx.x] += ssum[threadIdx.x + st];
            ssq[threadIdx.x]  += ssq[threadIdx.x + st];
        }
        __syncthreads();
    }
    if (threadIdx.x == 0) {
        const float mean = ssum[0] / (float)n;
        const float var  = ssq[0] / (float)n - mean * mean;  // biased variance
        const float sc   = gamma[f] * rsqrtf(var + BN_EPS);
        scale[f] = sc;
        shift[f] = beta[f] - mean * sc;
    }
}

// ---------------- BN apply (+optional ReLU) ----------------
__global__ void bn_apply(const float* __restrict__ agg, const float* __restrict__ scale,
                         const float* __restrict__ shift, float* __restrict__ h, int relu) {
    const int i = blockIdx.x;
    const int f = threadIdx.x;
    if (f < F) {
        float v = scale[f] * agg[(size_t)i * F + f] + shift[f];
        if (relu) v = v > 0.f ? v : 0.f;
        h[(size_t)i * F + f] = v;
    }
}

// ---------------- global add pool ----------------
__global__ void pool_kernel(const long long* __restrict__ batch, const float* __restrict__ h,
                            float* __restrict__ g) {
    const int i = blockIdx.x;
    const int f = threadIdx.x;
    if (f < F) {
        atomicAdd(&g[(size_t)batch[i] * F + f], h[(size_t)i * F + f]);
    }
}

extern "C" void kernel_launch(void* const* d_in, const int* in_sizes, int n_in,
                              void* d_out, int out_size, void* d_ws, size_t ws_size,
                              hipStream_t stream) {
    const float*     x     = (const float*)d_in[0];
    const long long* ei    = (const long long*)d_in[1];   // [2, NE] int64
    const long long* batch = (const long long*)d_in[2];   // [NN] int64
    const float*     W0    = (const float*)d_in[3];       // [33,100]
    const float*     Wrest = (const float*)d_in[4];       // [3,100,100]
    const float*     b     = (const float*)d_in[5];       // [4,100]
    const float*     gamma = (const float*)d_in[6];       // [4,100]
    const float*     beta  = (const float*)d_in[7];       // [4,100]
    const float*     Wout  = (const float*)d_in[8];       // [100,200]
    const float*     bout  = (const float*)d_in[9];       // [200]
    float* out = (float*)d_out;                           // [500,200]

    const long long* src = ei;
    const long long* dst = ei + NE;

    // workspace carve-out (256B aligned blocks)
    char* p = (char*)d_ws;
    auto carve = [&](size_t nfloats) {
        float* r = (float*)p;
        p += ((nfloats * sizeof(float) + 255) / 256) * 256;
        return r;
    };
    float* dinv  = carve(NN);                  // degree, then rsqrt in place
    float* xpad  = carve((size_t)NN * KPAD);   // zero-padded input (K=36)
    float* W0pad = carve((size_t)KPAD * F);    // zero-padded W0 rows (36x100)
    float* m     = carve((size_t)NN * F);      // h @ W
    float* agg   = carve((size_t)NN * F);      // scatter target
    float* h     = carve((size_t)NN * F);      // layer output
    float* g     = carve((size_t)NG * F);      // pooled
    float* scale = carve(F);
    float* shift = carve(F);

    // 1) GCN normalization (shared across layers)
    init_ones<<<(NN + 255) / 256, 256, 0, stream>>>(dinv, NN);
    deg_accum<<<(NE + 255) / 256, 256, 0, stream>>>(dst, dinv, NE);
    deg_to_dinv<<<(NN + 255) / 256, 256, 0, stream>>>(dinv, NN);

    // Pad x (K: 33 -> 36, zero tail) and W0 (rows 33 -> 36, zero tail):
    // zero tail of xpad contributes 0 regardless of W0pad tail, and vice versa.
    pad_x<<<NN, 64, 0, stream>>>(x, xpad);
    hipMemsetAsync(W0pad, 0, (size_t)KPAD * F * sizeof(float), stream);
    hipMemcpyAsync(W0pad, W0, (size_t)33 * F * sizeof(float),
                   hipMemcpyDeviceToDevice, stream);

    // 2) four GCN layers
    const float* hin = xpad;
    for (int layer = 0; layer < 4; ++layer) {
        const int K = (layer == 0) ? KPAD : F;
        const float* W = (layer == 0) ? W0pad : (Wrest + (size_t)(layer - 1) * F * F);

        const int tiles = ((NN + 15) / 16) * ((F + 15) / 16);
        wmma_gemm_f32<<<(tiles + 7) / 8, 256, 0, stream>>>(hin, W, m, NN, F, K, nullptr, 0);

        hipMemsetAsync(agg, 0, (size_t)NN * F * sizeof(float), stream);
        edge_scatter<<<(NE + 3) / 4, 128, 0, stream>>>(src, dst, dinv, m, agg, NE);
        selfloop_bias<<<NN, 128, 0, stream>>>(dinv, m, b + layer * F, agg);

        bn_stats<<<F, 256, 0, stream>>>(agg, gamma + layer * F, beta + layer * F,
                                        scale, shift, NN);
        bn_apply<<<NN, 128, 0, stream>>>(agg, scale, shift, h, layer < 3 ? 1 : 0);
        hin = h;
    }

    // 3) pool + output head (bias + LeakyReLU 0.1 fused into WMMA epilogue)
    hipMemsetAsync(g, 0, (size_t)NG * F * sizeof(float), stream);
    pool_kernel<<<NN, 128, 0, stream>>>(batch, h, g);

    const int tiles2 = ((NG + 15) / 16) * ((200 + 15) / 16);
    wmma_gemm_f32<<<(tiles2 + 7) / 8, 256, 0, stream>>>(g, Wout, out, NG, 200, F, bout, 1);
}